// RnnForWeight_23484881175133
// MI455X (gfx1250) — compile-verified
//
#include <hip/hip_runtime.h>
#include <cmath>

typedef __attribute__((ext_vector_type(2))) float v2f;
typedef __attribute__((ext_vector_type(8))) float v8f;

#define SEQ   64
#define INSZ  100000
#define HID   50
#define G4    200     // 4*H
#define NPAD  208     // 13 N-tiles * 16
#define NTIL  13

// ---------------------------------------------------------------------------
// Phase 1: split-K GEMM  part[chunk][64][208] = x[64, Kc] @ Wi[Kc, 200]
// One workgroup = one K chunk; 13 waves, one N-tile (16 cols) per wave.
// Each wave holds 4 M-tile accumulators (M = 64 = 4 x 16) and advances K by 4
// per v_wmma_f32_16x16x4_f32. Wi is streamed once -> non-temporal loads.
// Padding columns (>=200) are clamped (not zeroed) to keep the loop
// branch-free; their results land in the padding and are never read.
// ---------------------------------------------------------------------------
__global__ __launch_bounds__(416) void gemm_split_k(
        const float* __restrict__ x,
        const float* __restrict__ Wi,
        float* __restrict__ part, int KC) {
    const int lane  = threadIdx.x & 31;
    const int wave  = threadIdx.x >> 5;        // 0..12  -> N tile
    const int chunk = blockIdx.x;
    const int k0    = chunk * KC;
    const int col   = wave * 16 + (lane & 15); // 0..207 (>=200 padded)
    const int colc  = col < G4 ? col : G4 - 1; // clamped, always in-bounds
    const int mrow  = lane & 15;
    const int hi    = lane >> 4;               // lane half selects K 0/1 vs 2/3

    v8f acc[4] = {};                           // 4 M tiles of 16x16 f32

    // B pointer walks rows kk, kk+1 of Wi at column colc; +4 rows per K-step.
    const float* wp = Wi + (size_t)(k0 + (hi << 1)) * G4 + colc;
    const float* xp = x + (size_t)mrow * INSZ + k0 + (hi << 1);

    const int steps = KC >> 2;
#pragma unroll 2
    for (int it = 0; it < steps; ++it) {
        v2f B = { __builtin_nontemporal_load(wp),
                  __builtin_nontemporal_load(wp + G4) };
        wp += 4 * G4;
#pragma unroll
        for (int mt = 0; mt < 4; ++mt) {
            // A tile: x[mt*16+mrow][kk..kk+1] (8B aligned: kk even, INSZ even)
            v2f A = *(const v2f*)(xp + (size_t)mt * 16 * INSZ + 4 * it);
            acc[mt] = __builtin_amdgcn_wmma_f32_16x16x4_f32(
                false, A, false, B, (short)0, acc[mt], false, false);
        }
    }

    // D layout: lane half 0 -> M = r, lane half 1 -> M = r+8; N = col
    float* p = part + (size_t)chunk * SEQ * NPAD;
#pragma unroll
    for (int mt = 0; mt < 4; ++mt) {
#pragma unroll
        for (int r = 0; r < 8; ++r) {
            const int row = mt * 16 + r + (hi << 3);
            p[row * NPAD + col] = acc[mt][r];
        }
    }
}

// ---------------------------------------------------------------------------
// Phase 2: deterministic split-K reduction + (bi + bh)
// ---------------------------------------------------------------------------
__global__ void reduce_bias(const float* __restrict__ part,
                            const float* __restrict__ bi,
                            const float* __restrict__ bh,
                            float* __restrict__ gates, int chunks) {
    const int idx = blockIdx.x * blockDim.x + threadIdx.x;
    if (idx >= SEQ * G4) return;
    const int s = idx / G4;
    const int g = idx % G4;
    float sum = bi[g] + bh[g];
    for (int c = 0; c < chunks; ++c)
        sum += part[(size_t)c * SEQ * NPAD + s * NPAD + g];
    gates[idx] = sum;
}

// ---------------------------------------------------------------------------
// Phase 3: 64-step LSTM recurrence, one workgroup.
// Thread t (<200) owns gate column t with its 50 Wh weights in registers.
// h broadcast via LDS; c lives in registers of threads 0..49.
// Gate order matches jnp.split: [0:50)=i, [50:100)=f, [100:150)=g, [150:200)=o
// ---------------------------------------------------------------------------
__global__ __launch_bounds__(256) void lstm_seq(
        const float* __restrict__ gates,
        const float* __restrict__ Wh,
        float* __restrict__ out) {
    const int t = threadIdx.x;
    __shared__ float h[HID];
    __shared__ float ga[G4];

    float wr[HID];
    if (t < G4) {
#pragma unroll
        for (int j = 0; j < HID; ++j) wr[j] = Wh[j * G4 + t];
    }
    if (t < HID) h[t] = 0.f;
    float c = 0.f;
    __syncthreads();

    for (int s = 0; s < SEQ; ++s) {
        if (t < G4) {
            float sum = gates[s * G4 + t];
#pragma unroll
            for (int j = 0; j < HID; ++j) sum += h[j] * wr[j];
            float a;
            if (t >= 2 * HID && t < 3 * HID) a = tanhf(sum);        // g gate
            else                             a = 1.f / (1.f + __expf(-sum));
            ga[t] = a;
        }
        __syncthreads();
        if (t < HID) {
            c = ga[HID + t] * c + ga[t] * ga[2 * HID + t];
            h[t] = ga[3 * HID + t] * tanhf(c);
        }
        __syncthreads();
    }
    if (t < HID) out[t] = h[t];
}

// ---------------------------------------------------------------------------
extern "C" void kernel_launch(void* const* d_in, const int* in_sizes, int n_in,
                              void* d_out, int out_size, void* d_ws, size_t ws_size,
                              hipStream_t stream) {
    const float* x  = (const float*)d_in[0];
    const float* Wi = (const float*)d_in[1];
    const float* bi = (const float*)d_in[2];
    const float* Wh = (const float*)d_in[3];
    const float* bh = (const float*)d_in[4];
    float* out = (float*)d_out;

    // Pick split-K chunk count by available workspace (all divide 100000,
    // KC divisible by 4). More chunks -> more HBM-level parallelism.
    auto need = [](int c) {
        return ((size_t)c * SEQ * NPAD + (size_t)SEQ * G4) * sizeof(float);
    };
    int chunks = 200;
    if (ws_size < need(200)) chunks = 100;
    if (ws_size < need(100)) chunks = 50;
    if (ws_size < need(50))  chunks = 25;
    const int KC = INSZ / chunks;

    float* part  = (float*)d_ws;
    float* gates = part + (size_t)chunks * SEQ * NPAD;

    gemm_split_k<<<chunks, 416, 0, stream>>>(x, Wi, part, KC);
    reduce_bias<<<(SEQ * G4 + 255) / 256, 256, 0, stream>>>(part, bi, bh, gates, chunks);
    lstm_seq<<<1, 256, 0, stream>>>(gates, Wh, out);
}